// AvatarGaussianEstimator_72662256714012
// MI455X (gfx1250) — compile-verified
//
#include <hip/hip_runtime.h>
#include <stdint.h>

// Problem constants fixed by the reference setup.
constexpr int B_  = 4;
constexpr int C_  = 128;
constexpr int C4  = C_ / 4;   // channels in float4 units
constexpr int HF  = 128;      // feature map H
constexpr int WF  = 128;      // feature map W
constexpr int IMH = 512;      // img_h
constexpr int IMW = 512;      // img_w
constexpr int HW  = IMH * IMW;

typedef float f32x4 __attribute__((ext_vector_type(4)));

// ---------------------------------------------------------------------------
// Kernel 1: initialize per-batch depth maps to +inf (b128 stores).
// ---------------------------------------------------------------------------
__global__ void k_init_depth(f32x4* __restrict__ depth4, int total4) {
  int i = blockIdx.x * blockDim.x + threadIdx.x;
  if (i < total4) {
    const float inf = __int_as_float(0x7F800000);
    f32x4 v = {inf, inf, inf, inf};
    depth4[i] = v;
  }
}

// ---------------------------------------------------------------------------
// Kernel 2: scatter-min vertex depths into the depth maps.
// z in [1,5] > 0, so uint bit-pattern ordering == float ordering.
// ---------------------------------------------------------------------------
__global__ void k_depth_scatter(const float* __restrict__ v2d,
                                const float* __restrict__ v3d,
                                unsigned int* __restrict__ depth_u,
                                int Nv) {
  int i = blockIdx.x * blockDim.x + threadIdx.x;
  int total = B_ * Nv;
  if (i >= total) return;
  int b = i / Nv;
  int n = i - b * Nv;
  float2 xy = ((const float2*)v2d)[(size_t)b * Nv + n];
  float z   = v3d[((size_t)b * Nv + n) * 3 + 2];
  int xi = (int)rintf(xy.x);                 // rintf == round-half-even (jnp.round)
  int yi = (int)rintf(xy.y);
  if (xi >= 0 && xi < IMW && yi >= 0 && yi < IMH) {
    atomicMin(depth_u + (size_t)b * HW + (size_t)yi * IMW + xi,
              __float_as_uint(z));
  }
}

// ---------------------------------------------------------------------------
// Kernel 3: layout transform feature_map (B,C,H,W) -> (B,H,W,C).
// Loads are staged global->LDS with the CDNA5 async-to-LDS path
// (GLOBAL_LOAD_ASYNC_TO_LDS_B32, ASYNCcnt + s_wait_asynccnt), then written
// out transposed and coalesced. 32x32 tile, 33-float row pad keeps the
// transposed LDS reads bank-conflict-free (stride 33 mod 64 banks).
// ---------------------------------------------------------------------------
__global__ void k_transpose_fm(const float* __restrict__ fm,
                               float* __restrict__ fmT) {
  __shared__ float tile[32][33];
  const int b  = blockIdx.z;
  const int y  = blockIdx.y;
  const int x0 = blockIdx.x * 32;
  const int t  = threadIdx.x;        // 0..255 (8 waves of wave32)
  const int xx = t & 31;             // load mapping: lane sweeps x (coalesced)
  const int cL = t >> 5;             // 8 channel-rows per pass
  const int cc = t & 31;             // store mapping: lane sweeps c (coalesced)
  const int xL = t >> 5;

  for (int c0 = 0; c0 < C_; c0 += 32) {
    // ---- async load 32(c) x 32(x) tile into LDS -------------------------
    for (int i = 0; i < 4; ++i) {
      int c = c0 + cL + i * 8;
      uint32_t goff = (uint32_t)((((b * C_ + c) * HF + y) * WF + x0 + xx) * 4);
      uint32_t loff = (uint32_t)(uintptr_t)&tile[cL + i * 8][xx];
      // GVS mode: mem = SGPR64(base) + VGPR32(byte off); dest LDS addr in VGPR.
      asm volatile("global_load_async_to_lds_b32 %0, %1, %2"
                   :: "v"(loff), "v"(goff), "s"(fm) : "memory");
    }
    asm volatile("s_wait_asynccnt 0" ::: "memory");
    __syncthreads();
    // ---- write out, contiguous in channel -------------------------------
    for (int i = 0; i < 4; ++i) {
      int x = x0 + xL + i * 8;
      fmT[(((size_t)b * HF + y) * WF + x) * C_ + c0 + cc] = tile[cc][xL + i * 8];
    }
    __syncthreads();
  }
}

// ---------------------------------------------------------------------------
// Kernel 4: main per-gaussian kernel.
// Phase 1: each of 128 threads computes one gaussian's setup (centers,
//          normal/angle weight, visibility, bilinear weights + float4-unit
//          corner offsets), writes view_weights / centers3d, parks the
//          sampling params in LDS.
// Phase 2: wave-per-gaussian, lane = 4 channels: 4 coalesced b128 corner
//          gathers from channel-last fmT (L2-resident) + one coalesced
//          512B non-temporal b128 store per gaussian. Waves stride the
//          block's gaussians independently (no barrier in the loop).
// ---------------------------------------------------------------------------
__global__ void __launch_bounds__(128)
k_main(const float* __restrict__ fmT,
       const float* __restrict__ depth,
       const float* __restrict__ v2d,
       const float* __restrict__ v3d,
       const int*   __restrict__ parents,
       const float* __restrict__ bary,
       float* __restrict__ out_feats,
       float* __restrict__ out_vw,
       float* __restrict__ out_c3,
       int N, int Nv, int K) {
  __shared__ float s_w[128][4];
  __shared__ int   s_o[128][4];   // corner offsets in float4 units

  const int t    = threadIdx.x;   // 0..127 (4 waves)
  const int b    = blockIdx.y;
  const int base = blockIdx.x * 128;
  const int n    = base + t;

  if (n < N) {
    // --- gather parents + barycentric weights ---------------------------
    int p0 = parents[n * 3 + 0];
    int p1 = parents[n * 3 + 1];
    int p2 = parents[n * 3 + 2];
    int kb = n % K;
    float w0 = bary[kb * 3 + 0];
    float w1 = bary[kb * 3 + 1];
    float w2 = bary[kb * 3 + 2];

    // --- 2D centers ------------------------------------------------------
    const float2* v2 = (const float2*)v2d;
    size_t vb = (size_t)b * Nv;
    float2 A2 = v2[vb + p0], B2 = v2[vb + p1], C2 = v2[vb + p2];
    float cx = w0 * A2.x + w1 * B2.x + w2 * C2.x;
    float cy = w0 * A2.y + w1 * B2.y + w2 * C2.y;

    // --- 3D centers + face normal ---------------------------------------
    const float* v3 = v3d + vb * 3;
    float ax = v3[p0 * 3 + 0], ay = v3[p0 * 3 + 1], az = v3[p0 * 3 + 2];
    float bx = v3[p1 * 3 + 0], by = v3[p1 * 3 + 1], bz = v3[p1 * 3 + 2];
    float gx = v3[p2 * 3 + 0], gy = v3[p2 * 3 + 1], gz = v3[p2 * 3 + 2];
    float c3x = w0 * ax + w1 * bx + w2 * gx;
    float c3y = w0 * ay + w1 * by + w2 * gy;
    float c3z = w0 * az + w1 * bz + w2 * gz;

    float e1x = bx - ax, e1y = by - ay, e1z = bz - az;
    float e2x = gx - ax, e2y = gy - ay, e2z = gz - az;
    float nx = e1y * e2z - e1z * e2y;
    float ny = e1z * e2x - e1x * e2z;
    float nz = e1x * e2y - e1y * e2x;
    float nl = sqrtf(nx * nx + ny * ny + nz * nz) + 1e-8f;
    float vl = sqrtf(c3x * c3x + c3y * c3y + c3z * c3z) + 1e-8f;
    float aw = fmaxf(-(nx * c3x + ny * c3y + nz * c3z) / (nl * vl), 0.0f);

    // --- visibility via depth map ---------------------------------------
    int xc = min(max((int)rintf(cx), 0), IMW - 1);
    int yc = min(max((int)rintf(cy), 0), IMH - 1);
    float dsamp = depth[(size_t)b * HW + (size_t)yc * IMW + xc];
    // dsamp==+inf (empty pixel) also passes <=, matching the reference.
    float vw = (c3z <= dsamp + 1e-3f) ? aw : 0.0f;
    out_vw[(size_t)b * N + n] = vw;
    size_t c3o = ((size_t)b * N + n) * 3;
    out_c3[c3o + 0] = c3x;
    out_c3[c3o + 1] = c3y;
    out_c3[c3o + 2] = c3z;

    // --- bilinear sampling setup ----------------------------------------
    // px = cx/(W-1)*(Wf-1), py = cy/(H-1)*(Hf-1)
    float px = cx * ((float)(WF - 1) / (float)(IMW - 1));
    float py = cy * ((float)(HF - 1) / (float)(IMH - 1));
    float x0f = floorf(px), y0f = floorf(py);
    float wx = px - x0f, wy = py - y0f;
    int x0 = (int)x0f, y0 = (int)y0f;
    int x1 = x0 + 1,   y1 = y0 + 1;
    float okx0 = (x0 >= 0 && x0 < WF) ? 1.0f : 0.0f;
    float okx1 = (x1 >= 0 && x1 < WF) ? 1.0f : 0.0f;
    float oky0 = (y0 >= 0 && y0 < HF) ? 1.0f : 0.0f;
    float oky1 = (y1 >= 0 && y1 < HF) ? 1.0f : 0.0f;
    int x0c = min(max(x0, 0), WF - 1), x1c = min(max(x1, 0), WF - 1);
    int y0c = min(max(y0, 0), HF - 1), y1c = min(max(y1, 0), HF - 1);
    int rowb = b * HF;
    s_w[t][0] = (1.0f - wx) * (1.0f - wy) * okx0 * oky0;
    s_w[t][1] = wx * (1.0f - wy) * okx1 * oky0;
    s_w[t][2] = (1.0f - wx) * wy * okx0 * oky1;
    s_w[t][3] = wx * wy * okx1 * oky1;
    s_o[t][0] = ((rowb + y0c) * WF + x0c) * C4;   // float4-unit offsets
    s_o[t][1] = ((rowb + y0c) * WF + x1c) * C4;
    s_o[t][2] = ((rowb + y1c) * WF + x0c) * C4;
    s_o[t][3] = ((rowb + y1c) * WF + x1c) * C4;
  }
  __syncthreads();

  // --- phase 2: wave-per-gaussian, float4 per lane ------------------------
  const int wave = t >> 5;        // 0..3
  const int lane = t & 31;        // lane -> channels [4*lane, 4*lane+3]
  const int cnt  = min(128, N - base);
  const f32x4* fmT4      = (const f32x4*)fmT;
  f32x4*       out4      = (f32x4*)out_feats;

  for (int i = wave; i < cnt; i += 4) {
    if (i + 4 < cnt) {  // prefetch next gaussian's corners (global_prefetch_b8)
      __builtin_prefetch(fmT4 + s_o[i + 4][0] + lane, 0, 1);
      __builtin_prefetch(fmT4 + s_o[i + 4][3] + lane, 0, 1);
    }
    float w0 = s_w[i][0], w1 = s_w[i][1], w2 = s_w[i][2], w3 = s_w[i][3];
    f32x4 f0 = fmT4[s_o[i][0] + lane];
    f32x4 f1 = fmT4[s_o[i][1] + lane];
    f32x4 f2 = fmT4[s_o[i][2] + lane];
    f32x4 f3 = fmT4[s_o[i][3] + lane];
    f32x4 v;
    v.x = w0 * f0.x + w1 * f1.x + w2 * f2.x + w3 * f3.x;
    v.y = w0 * f0.y + w1 * f1.y + w2 * f2.y + w3 * f3.y;
    v.z = w0 * f0.z + w1 * f1.z + w2 * f2.z + w3 * f3.z;
    v.w = w0 * f0.w + w1 * f1.w + w2 * f2.w + w3 * f3.w;
    __builtin_nontemporal_store(
        v, out4 + ((size_t)b * N + (base + i)) * C4 + lane);
  }
}

// ---------------------------------------------------------------------------
// Host launcher. Workspace layout: [depth: B*HW floats][fmT: B*HF*WF*C floats]
// = 4 MB + 32 MB. Output layout (flat, return order): local_feats (B,N,C),
// view_weights (B,N), centers3d (B,N,3).
// ---------------------------------------------------------------------------
extern "C" void kernel_launch(void* const* d_in, const int* in_sizes, int n_in,
                              void* d_out, int out_size, void* d_ws, size_t ws_size,
                              hipStream_t stream) {
  const float* fm   = (const float*)d_in[0];
  const float* v2d  = (const float*)d_in[1];
  const float* v3d  = (const float*)d_in[2];
  const int*   par  = (const int*)d_in[3];
  const float* bary = (const float*)d_in[4];
  // d_in[5], d_in[6] = img_h, img_w (fixed at 512 by the reference setup)

  const int Nv = in_sizes[1] / (B_ * 2);
  const int N  = in_sizes[3] / 3;
  const int K  = in_sizes[4] / 3;

  float* depth = (float*)d_ws;
  float* fmT   = depth + (size_t)B_ * HW;

  float* out_feats = (float*)d_out;
  float* out_vw    = out_feats + (size_t)B_ * N * C_;
  float* out_c3    = out_vw    + (size_t)B_ * N;

  const int total4 = (B_ * HW) / 4;
  k_init_depth<<<(total4 + 255) / 256, 256, 0, stream>>>((f32x4*)depth, total4);

  k_depth_scatter<<<(B_ * Nv + 255) / 256, 256, 0, stream>>>(
      v2d, v3d, (unsigned int*)depth, Nv);

  k_transpose_fm<<<dim3(WF / 32, HF, B_), 256, 0, stream>>>(fm, fmT);

  k_main<<<dim3((N + 127) / 128, B_), 128, 0, stream>>>(
      fmT, depth, v2d, v3d, par, bary, out_feats, out_vw, out_c3, N, Nv, K);
}